// PSwater_45440753992042
// MI455X (gfx1250) — compile-verified
//
#include <hip/hip_runtime.h>
#include <stdint.h>

// ---------------------------------------------------------------------------
// PS-water energy/forces/dipole/BEC — MI455X (gfx1250, wave32)
//
// The sparse 243/82-term polynomials (and all their x-derivatives) are folded
// into three dense tensors W[k][i*16+j] (16x256 f32). Per wave, 16 molecules
// (one per C/D column, 2 lanes each) evaluate all nine reductions
//     u_k = sum_ij W[k][ij] * O[ij],   O in {F0xF1, G0xF1, F0xG1}
// as chained V_WMMA_F32_16X16X4_F32: A = shared W chunk from LDS (imm-offset
// ds_load_b64), B = O chunk built from register-resident power tables with
// static indices (i = c/4 is chunk-uniform). W is staged to LDS with
// global_load_async_to_lds_b128; outputs drain via global_store_async_from_lds.
// ---------------------------------------------------------------------------

typedef __attribute__((ext_vector_type(8))) float v8f;
typedef __attribute__((ext_vector_type(4))) float v4f;
typedef __attribute__((ext_vector_type(2))) float v2f;

#define NTERMS 244
#define NDIP   84
#define BLOCK  64
#define WROW   260                 // padded row stride (dwords): banks for b64
#define WTEN   (16*WROW)           // 4160 floats per tensor
#define WTOT   (3*WTEN)            // 12480 floats = 49920 B
#define STAGE_PER_WAVE 640         // 16 mol * (1+9+3+27)

// ---------------- kernel 1: build dense W tensors (deterministic) ----------
__global__ __launch_bounds__(BLOCK)
void build_W(const float* __restrict__ c5z,
             const float* __restrict__ coefD,
             const int*   __restrict__ idx,
             float*       __restrict__ Wg)
{
    __shared__ __align__(16) float sW[WTOT];
    const int tid = threadIdx.x;
    for (int e = tid; e < WTOT; e += BLOCK) sW[e] = 0.f;
    __syncthreads();
    // cell ownership by (i*16+j)&63 -> no write conflicts, deterministic order
    for (int t = 1; t < NTERMS; ++t) {
        int i = idx[3*t], j = idx[3*t+1], k = idx[3*t+2];
        float c = c5z[t];
        if (((i*16+j) & 63) == tid) sW[k*WROW + i*16+j] += c;   // symmetrized
        if (((j*16+i) & 63) == tid) sW[k*WROW + j*16+i] += c;
    }
    for (int t = 2; t < NDIP; ++t) {
        int i = idx[3*t], j = idx[3*t+1], k = idx[3*t+2];
        float c = coefD[t];
        if (((i*16+j) & 63) == tid) sW[WTEN   + k*WROW + i*16+j] += c; // P1
        if (((j*16+i) & 63) == tid) sW[2*WTEN + k*WROW + j*16+i] += c; // P2
    }
    __syncthreads();
    for (int e = tid; e < WTOT/4; e += BLOCK)
        *(v4f*)(Wg + 4*e) = *(const v4f*)(sW + 4*e);
}

// ---------------- kernel 2: main WMMA kernel -------------------------------
__global__ __launch_bounds__(BLOCK)
void pswater_wmma(const float* __restrict__ positions,
                  const float* __restrict__ c5z,
                  const float* __restrict__ coefD,
                  const float* __restrict__ cst,
                  const float* __restrict__ Wg,
                  float*       __restrict__ out,
                  int n)
{
    __shared__ __align__(16) float sW[WTOT];                // 49920 B
    __shared__ __align__(16) float sStage[2*STAGE_PER_WAVE];//  5120 B

    const int tid  = threadIdx.x;
    const int lane = tid & 31;
    const int wave = tid >> 5;
    const int lm   = lane & 15;        // molecule-in-wave (C/D column)
    const int half = lane >> 4;        // 0: rows 0-7 / K 0-1, 1: rows 8-15 / K 2-3
    if (blockIdx.x*32 + 32 > n) return;            // keep EXEC all-ones
    const int mol  = blockIdx.x*32 + wave*16 + lm;

    __builtin_prefetch(positions + 9*(size_t)mol, 0, 3);

    // ---- kick off async W -> LDS DMA (overlaps with geometry below) ----
    {
        unsigned l0 = (unsigned)(uintptr_t)sW;
        for (int e = tid; e < WTOT/4; e += BLOCK) {
            unsigned lb = l0 + (unsigned)e*16u;
            unsigned vo = (unsigned)e*16u;
            asm volatile("global_load_async_to_lds_b128 %0, %1, %2"
                         :: "v"(lb), "v"(vo), "s"((unsigned long long)(uintptr_t)Wg)
                         : "memory");
        }
    }

    // ------------------------- geometry -------------------------
    const float* P = positions + 9*(size_t)mol;
    float p0x=P[0], p0y=P[1], p0z=P[2];
    float p1x=P[3], p1y=P[4], p1z=P[5];
    float p2x=P[6], p2y=P[7], p2z=P[8];
    float ux=p1x-p0x, uy=p1y-p0y, uz=p1z-p0z;   // ROH1
    float vx=p2x-p0x, vy=p2y-p0y, vz=p2z-p0z;   // ROH2
    float wx=p1x-p2x, wy=p1y-p2y, wz=p1z-p2z;   // RHH
    float uu = ux*ux+uy*uy+uz*uz;
    float vv = vx*vx+vy*vy+vz*vz;
    float ww = wx*wx+wy*wy+wz*wz;
    float inv_d1 = __builtin_amdgcn_rsqf(uu);  float d1  = uu*inv_d1;
    float inv_d2 = __builtin_amdgcn_rsqf(vv);  float d2  = vv*inv_d2;
    float inv_dh = __builtin_amdgcn_rsqf(ww);  float dhh = ww*inv_dh;
    float a12   = inv_d1*inv_d2;
    float costh = (ux*vx+uy*vy+uz*vz)*a12;

    const float alphaoh=cst[0], roh=cst[1], deoh=cst[2], phh1=cst[3],
                phh2=cst[4],    reoh=cst[5], b1=cst[6],  costhe=cst[7];
    float inv_reoh = __builtin_amdgcn_rcpf(reoh);
    float e1v = __expf(-alphaoh*(d1-roh));
    float e2v = __expf(-alphaoh*(d2-roh));
    float Va  = deoh*(e1v*(e1v-2.f)+e2v*(e2v-2.f));
    float Vb  = phh1*__expf(-phh2*dhh);
    float t1=d1-reoh, t2=d2-reoh;
    float x0=t1*inv_reoh, x1=t2*inv_reoh, x2=costh-costhe;
    float efac = __expf(-b1*(t1*t1+t2*t2));

    // ---- register power tables: F(v)=x^(v-1) (F[0]=0), G(v)=dF/dx ----
    float F0[16],F1[16],F2[16],G0[16],G1[16],G2[16];
    F0[0]=0.f; F1[0]=0.f; F2[0]=0.f; G0[0]=0.f; G1[0]=0.f; G2[0]=0.f;
    F0[1]=1.f; F1[1]=1.f; F2[1]=1.f; G0[1]=0.f; G1[1]=0.f; G2[1]=0.f;
    #pragma unroll
    for (int v = 2; v < 16; ++v) {
        float cv = (float)(v-1);
        F0[v]=F0[v-1]*x0; G0[v]=cv*F0[v-1];
        F1[v]=F1[v-1]*x1; G1[v]=cv*F1[v-1];
        F2[v]=F2[v-1]*x2; G2[v]=cv*F2[v-1];
    }
    // per-half operand selects (hoisted out of the chunk loop)
    float f1a[4], f1b[4], g1a[4], g1b[4];
    #pragma unroll
    for (int q = 0; q < 4; ++q) {
        f1a[q] = half ? F1[4*q+2] : F1[4*q];
        f1b[q] = half ? F1[4*q+3] : F1[4*q+1];
        g1a[q] = half ? G1[4*q+2] : G1[4*q];
        g1b[q] = half ? G1[4*q+3] : G1[4*q+1];
    }
    float cs[8], gs[8];                      // C / C' rows for this lane half
    #pragma unroll
    for (int r = 0; r < 8; ++r) {
        cs[r] = half ? F2[r+8] : F2[r];
        gs[r] = half ? G2[r+8] : G2[r];
    }

    // ---- wait for W, then the WMMA contraction ----
    asm volatile("s_wait_asynccnt 0" ::: "memory");
    __syncthreads();

    const float* pW = sW + lm*WROW + half*2;          // A-operand lane base
    v8f aEO ={0.f,0.f,0.f,0.f,0.f,0.f,0.f,0.f}, aE1=aEO, aE2=aEO;
    v8f aD1O=aEO, aD11=aEO, aD12=aEO;
    v8f aD2O=aEO, aD21=aEO, aD22=aEO;
    #pragma unroll
    for (int c = 4; c < 64; ++c) {                    // c<4 => i=0 => B==0
        const int i = c >> 2, q = c & 3;
        v2f wE  = *(const v2f*)(pW          + 4*c);   // ds_load_b64, imm offs
        v2f wD1 = *(const v2f*)(pW + WTEN   + 4*c);
        v2f wD2 = *(const v2f*)(pW + 2*WTEN + 4*c);
        v2f bO  = { F0[i]*f1a[q], F0[i]*f1b[q] };     // O   = F0 x F1
        v2f bO1 = { G0[i]*f1a[q], G0[i]*f1b[q] };     // O'  = G0 x F1
        v2f bO2 = { F0[i]*g1a[q], F0[i]*g1b[q] };     // O'' = F0 x G1
        aEO  = __builtin_amdgcn_wmma_f32_16x16x4_f32(false, wE,  false, bO,  (short)0, aEO,  false, false);
        aE1  = __builtin_amdgcn_wmma_f32_16x16x4_f32(false, wE,  false, bO1, (short)0, aE1,  false, false);
        aE2  = __builtin_amdgcn_wmma_f32_16x16x4_f32(false, wE,  false, bO2, (short)0, aE2,  false, false);
        aD1O = __builtin_amdgcn_wmma_f32_16x16x4_f32(false, wD1, false, bO,  (short)0, aD1O, false, false);
        aD11 = __builtin_amdgcn_wmma_f32_16x16x4_f32(false, wD1, false, bO1, (short)0, aD11, false, false);
        aD12 = __builtin_amdgcn_wmma_f32_16x16x4_f32(false, wD1, false, bO2, (short)0, aD12, false, false);
        aD2O = __builtin_amdgcn_wmma_f32_16x16x4_f32(false, wD2, false, bO,  (short)0, aD2O, false, false);
        aD21 = __builtin_amdgcn_wmma_f32_16x16x4_f32(false, wD2, false, bO1, (short)0, aD21, false, false);
        aD22 = __builtin_amdgcn_wmma_f32_16x16x4_f32(false, wD2, false, bO2, (short)0, aD22, false, false);
    }

    // ---- dots with C (=F2) / C' (=G2); combine lane halves ----
    auto dot8 = [&](const v8f& a, const float* cv) -> float {
        float s = 0.f;
        #pragma unroll
        for (int r = 0; r < 8; ++r) s += a[r]*cv[r];
        return s;
    };
    auto red = [&](float x) -> float { return x + __shfl_xor(x, 16); };
    float s0  = red(dot8(aEO,  cs));
    float g0  = red(dot8(aE1,  cs));
    float g1  = red(dot8(aE2,  cs));
    float g2  = red(dot8(aEO,  gs));
    float P1s = red(dot8(aD1O, cs));
    float q10 = red(dot8(aD11, cs));
    float q11 = red(dot8(aD12, cs));
    float q12 = red(dot8(aD1O, gs));
    float P2s = red(dot8(aD2O, cs));
    float q20 = red(dot8(aD21, cs));
    float q21 = red(dot8(aD22, cs));
    float q22 = red(dot8(aD2O, gs));

    // ------------------------- epilogue (per lane) -------------------------
    const float EK = 0.002859143751006429f;
    float Vc = 2.f*c5z[0] + efac*s0;
    float energy = (Va+Vb+Vc+0.44739574026257f)*EK;
    float dVa1 = -2.f*alphaoh*deoh*e1v*(e1v-1.f);
    float dVa2 = -2.f*alphaoh*deoh*e2v*(e2v-1.f);
    float def1 = -2.f*b1*t1*efac;
    float def2 = -2.f*b1*t2*efac;
    float dE1 = EK*(dVa1 + def1*s0 + efac*g0*inv_reoh);
    float dE2 = EK*(dVa2 + def2*s0 + efac*g1*inv_reoh);
    float dEh = EK*(-phh2*Vb);
    float dEc = EK*(efac*g2);

    float un[3]={ux*inv_d1,uy*inv_d1,uz*inv_d1};
    float vn[3]={vx*inv_d2,vy*inv_d2,vz*inv_d2};
    float wn[3]={wx*inv_dh,wy*inv_dh,wz*inv_dh};
    float i1s=inv_d1*inv_d1, i2s=inv_d2*inv_d2;
    float uvec[3]={ux,uy,uz}, vvec[3]={vx,vy,vz};
    float dcu[3], dcv[3];
    #pragma unroll
    for (int b = 0; b < 3; ++b) {
        dcu[b] = vvec[b]*a12 - costh*uvec[b]*i1s;
        dcv[b] = uvec[b]*a12 - costh*vvec[b]*i2s;
    }
    float F[3][3];
    #pragma unroll
    for (int b = 0; b < 3; ++b) {
        F[0][b] =  dE1*un[b] + dE2*vn[b] + dEc*(dcu[b]+dcv[b]);
        F[1][b] = -(dE1*un[b] + dEh*wn[b] + dEc*dcu[b]);
        F[2][b] = -(dE2*vn[b] - dEh*wn[b] + dEc*dcv[b]);
    }

    const float PEXP = -0.6932f, CC = 0.529177249f;
    float pa = __powf(d1, PEXP), pb = __powf(d2, PEXP);
    float gg    = 1.0099f - 0.1801f*costh + 0.0446f*(3.f*costh*costh - 1.f);
    float sab   = 0.2999f*(pa+pb);
    float PC0   = sab*gg;
    float dPC01 = 0.2999f*PEXP*pa*inv_d1*gg;
    float dPC02 = 0.2999f*PEXP*pb*inv_d2*gg;
    float dPC0c = sab*(-0.1801f + 0.2676f*costh);
    float cd0 = coefD[0];
    float P1 = cd0 + P1s*efac + PC0*CC;
    float P2 = cd0 + P2s*efac + PC0*CC;
    float dP1_1 = q10*inv_reoh*efac + P1s*def1 + CC*dPC01;
    float dP1_2 = q11*inv_reoh*efac + P1s*def2 + CC*dPC02;
    float dP1_c = q12*efac + CC*dPC0c;
    float dP2_1 = q20*inv_reoh*efac + P2s*def1 + CC*dPC01;
    float dP2_2 = q21*inv_reoh*efac + P2s*def2 + CC*dPC02;
    float dP2_c = q22*efac + CC*dPC0c;

    float qv[3] = {-(P1+P2), P1, P2};
    float pos0[3]={p0x,p0y,p0z}, pos1[3]={p1x,p1y,p1z}, pos2[3]={p2x,p2y,p2z};
    float dq0_1=-(dP1_1+dP2_1), dq0_2=-(dP1_2+dP2_2), dq0_c=-(dP1_c+dP2_c);
    float dip[3], Gd1[3], Gd2[3], Gdc[3];
    #pragma unroll
    for (int a = 0; a < 3; ++a) {
        dip[a] = qv[0]*pos0[a] + qv[1]*pos1[a] + qv[2]*pos2[a];
        Gd1[a] = pos0[a]*dq0_1 + pos1[a]*dP1_1 + pos2[a]*dP2_1;
        Gd2[a] = pos0[a]*dq0_2 + pos1[a]*dP1_2 + pos2[a]*dP2_2;
        Gdc[a] = pos0[a]*dq0_c + pos1[a]*dP1_c + pos2[a]*dP2_c;
    }
    float D1f[3][3], D2f[3][3], DCf[3][3];
    #pragma unroll
    for (int b = 0; b < 3; ++b) {
        D1f[0][b]=-un[b]; D1f[1][b]= un[b]; D1f[2][b]=0.f;
        D2f[0][b]=-vn[b]; D2f[1][b]= 0.f;   D2f[2][b]=vn[b];
        DCf[0][b]=-(dcu[b]+dcv[b]); DCf[1][b]=dcu[b]; DCf[2][b]=dcv[b];
    }

    // ---- stage (flat global order, lanes<16 own the 16 molecules) ----
    float* SW_ = sStage + wave*STAGE_PER_WAVE;
    float* SE = SW_;            // 16
    float* SF = SW_ + 16;       // 144
    float* SD = SW_ + 160;      // 48
    float* SB = SW_ + 208;      // 432
    if (!half) {
        SE[lm] = energy;
        #pragma unroll
        for (int j = 0; j < 3; ++j)
            #pragma unroll
            for (int b = 0; b < 3; ++b)
                SF[lm*9 + j*3 + b] = F[j][b];
        #pragma unroll
        for (int a = 0; a < 3; ++a) SD[lm*3 + a] = dip[a];
        #pragma unroll
        for (int j = 0; j < 3; ++j)
            #pragma unroll
            for (int b = 0; b < 3; ++b)
                #pragma unroll
                for (int a = 0; a < 3; ++a)
                    SB[lm*27 + (j*3+b)*3 + a] =
                        Gd1[a]*D1f[j][b] + Gd2[a]*D2f[j][b] + Gdc[a]*DCf[j][b]
                        + ((a == b) ? qv[j] : 0.f);
    }

    // ---- drain with async store-from-LDS DMA (coalesced) ----
    asm volatile("s_wait_dscnt 0" ::: "memory");
    const int waveFirst = blockIdx.x*32 + wave*16;
    auto async_out = [&](const float* gdst, const float* lsrc, int count) {
        unsigned l0 = (unsigned)(uintptr_t)lsrc;
        unsigned long long g = (unsigned long long)(uintptr_t)gdst;
        for (int e = lane; e < count; e += 32) {
            unsigned lb = l0 + (unsigned)e*4u;
            unsigned vo = (unsigned)e*4u;
            asm volatile("global_store_async_from_lds_b32 %0, %1, %2"
                         :: "v"(vo), "v"(lb), "s"(g) : "memory");
        }
    };
    async_out(out +                  (size_t)waveFirst,      SE,  16);
    async_out(out + (size_t)n      + (size_t)waveFirst*9,    SF, 144);
    async_out(out + (size_t)n*10   + (size_t)waveFirst*3,    SD,  48);
    async_out(out + (size_t)n*13   + (size_t)waveFirst*27,   SB, 432);
    asm volatile("s_wait_asynccnt 0" ::: "memory");
}

extern "C" void kernel_launch(void* const* d_in, const int* in_sizes, int n_in,
                              void* d_out, int out_size, void* d_ws, size_t ws_size,
                              hipStream_t stream) {
    const float* positions = (const float*)d_in[0];
    const float* c5z       = (const float*)d_in[1];
    const float* coefD     = (const float*)d_in[2];
    const float* consts    = (const float*)d_in[3];
    const int*   idx       = (const int*)  d_in[4];
    float* out = (float*)d_out;
    float* Wg  = (float*)d_ws;               // 49920 B of scratch
    int n = in_sizes[0] / 9;

    build_W<<<dim3(1), dim3(BLOCK), 0, stream>>>(c5z, coefD, idx, Wg);
    int blocks = (n + 31) / 32;
    pswater_wmma<<<dim3(blocks), dim3(BLOCK), 0, stream>>>(
        positions, c5z, coefD, consts, Wg, out, n);
}